// GraphVisCommNet_29703993819532
// MI455X (gfx1250) — compile-verified
//
#include <hip/hip_runtime.h>
#include <hip/hip_bf16.h>
#include <stdint.h>

// ---------------------------------------------------------------------------
// GraphVisCommNet on MI455X (gfx1250).
// Dense GEMMs: V_WMMA_F32_16X16X4_F32 (exact fp32; ~12 GFLOP, not bottleneck).
// B panels staged into LDS via gfx1250 async copy (global_load_async_to_lds_b128
// + s_wait_asynccnt); A fragments preloaded per K-chunk; chunk body is a
// branch-free compile-time-unrolled WMMA run. Edge softmax/scatter passes are
// the ~1.4 GB HBM-bound part (~60 us at 23.3 TB/s) and dominate runtime.
// ---------------------------------------------------------------------------

typedef __attribute__((ext_vector_type(2))) float v2f;
typedef __attribute__((ext_vector_type(8))) float v8f;

#define H4 4
#define N_GT 100000
#define N_AG 20000
#define E_GT 640000
#define E_COMM 320000
#define D_GT 32
#define D_AG 64
#define VIS 64
#define COMM 128
#define MSG 64
#define OUTD 5

#define KCH 64   // K chunk staged per iteration
#define BST 72   // LDS row stride in floats: half-groups hit disjoint bank sets

static inline unsigned cdiv(unsigned a, unsigned b) { return (a + b - 1) / b; }

// ---------------------------------------------------------------------------
// Branch-free chunk body: NF fragments (NF*4 K-depth), 4 N-subtiles.
// ---------------------------------------------------------------------------
template <int NF>
__device__ __forceinline__ void gemm_chunk(const float* __restrict__ ap,
                                           const float (&Bs)[KCH * BST],
                                           int half, int l16, v8f (&acc)[4])
{
    v2f afrag[NF];
#pragma unroll
    for (int f = 0; f < NF; ++f) {
        afrag[f].x = ap[4 * f];
        afrag[f].y = ap[4 * f + 1];
    }
#pragma unroll
    for (int f = 0; f < NF; ++f) {
        const int kk = f << 2;
#pragma unroll
        for (int t = 0; t < 4; ++t) {
            v2f b;
            b.x = Bs[(kk + 2 * half)     * BST + t * 16 + l16];
            b.y = Bs[(kk + 2 * half + 1) * BST + t * 16 + l16];
            acc[t] = __builtin_amdgcn_wmma_f32_16x16x4_f32(
                false, afrag[f], false, b, (short)0, acc[t], false, false);
        }
    }
}

// ---------------------------------------------------------------------------
// WMMA FP32 GEMM: C[M,N] (+)= A[M,K] @ B[K,N]  (+bias) (+relu), row-major.
// Block = 4 waves = 64 M-rows x 64 N-cols. Per wave: 4 16x16 accumulators.
// Requires: K % 4 == 0, N % 64 == 0, M % 16 == 0, K chunks in {32, 64}.
// ---------------------------------------------------------------------------
__global__ void gemm_wmma_f32(const float* __restrict__ A, int lda,
                              const float* __restrict__ B, int ldb,
                              float* __restrict__ C, int ldc,
                              int M, int N, int K,
                              const float* __restrict__ bias,
                              int accumulate, int do_relu)
{
    __shared__ float Bs[KCH * BST];

    const int tid  = threadIdx.x;
    const int wave = tid >> 5;
    const int lane = tid & 31;
    const int half = lane >> 4;   // 0 or 1
    const int l16  = lane & 15;
    const int m0 = (blockIdx.x * 4 + wave) * 16;
    const int n0 = blockIdx.y * 64;
    const bool active = (m0 < M);   // wave-uniform; inactive waves still barrier

    v8f acc[4];
#pragma unroll
    for (int t = 0; t < 4; ++t) acc[t] = (v8f){0.f,0.f,0.f,0.f,0.f,0.f,0.f,0.f};

    if (active && accumulate) {
#pragma unroll
        for (int t = 0; t < 4; ++t) {
            const float* cp = C + (size_t)(m0 + half * 8) * ldc + n0 + t * 16 + l16;
#pragma unroll
            for (int r = 0; r < 8; ++r) acc[t][r] = cp[(size_t)r * ldc];
        }
    }

    const unsigned bsBase = (unsigned)(uintptr_t)(&Bs[0]);  // LDS byte offset

    for (int kb = 0; kb < K; kb += KCH) {
        const int chunk = (K - kb < KCH) ? (K - kb) : KCH;  // 32 or 64

        __syncthreads();   // previous chunk's LDS reads complete before overwrite

        // --- async DMA: B[kb:kb+chunk, n0:n0+64] -> LDS (row stride BST) ---
        // chunk*16 b128 transfers; chunk*16 is a multiple of 128 for our shapes.
        for (int q = tid; q < chunk * 16; q += 128) {
            int kk = q >> 4;
            int c4 = (q & 15) << 2;
            const float* gp = B + (size_t)(kb + kk) * ldb + n0 + c4;
            unsigned lds = bsBase + (unsigned)((kk * BST + c4) * 4);
            asm volatile("global_load_async_to_lds_b128 %0, %1, off"
                         :: "v"(lds), "v"((unsigned long long)(uintptr_t)gp)
                         : "memory");
        }
        asm volatile("s_wait_asynccnt 0x0" ::: "memory");
        __syncthreads();

        if (active) {
            const float* ap = A + (size_t)(m0 + l16) * lda + kb + 2 * half;
            if (chunk == KCH) gemm_chunk<KCH / 4>(ap, Bs, half, l16, acc);
            else              gemm_chunk<8>(ap, Bs, half, l16, acc);  // chunk==32
        }
    }

    if (active) {
#pragma unroll
        for (int t = 0; t < 4; ++t) {
            float bv = bias ? bias[n0 + t * 16 + l16] : 0.0f;
            float* cp = C + (size_t)(m0 + half * 8) * ldc + n0 + t * 16 + l16;
#pragma unroll
            for (int r = 0; r < 8; ++r) {
                float v = acc[t][r] + bv;
                if (do_relu) v = v > 0.f ? v : 0.f;
                cp[(size_t)r * ldc] = v;
            }
        }
    }
}

// ---------------------------------------------------------------------------
// Elementwise / edge kernels
// ---------------------------------------------------------------------------
__global__ void fill_kernel(float* __restrict__ p, float v, int n)
{
    int t = blockIdx.x * blockDim.x + threadIdx.x;
    if (t < n) p[t] = v;
}

__device__ inline void atomicMaxF32(float* addr, float val)
{
    unsigned* ua = (unsigned*)addr;
    unsigned old = *ua;
    while (__uint_as_float(old) < val) {
        unsigned assumed = old;
        old = atomicCAS(ua, assumed, __float_as_uint(val));
        if (old == assumed) break;
    }
}

// el[i,h] = dot(feat[i, h*dh : (h+1)*dh], aw[h, :])
__global__ void rowdot_kernel(const float* __restrict__ feat,
                              const float* __restrict__ aw,
                              float* __restrict__ out, int n, int dh)
{
    int t = blockIdx.x * blockDim.x + threadIdx.x;
    if (t >= n * H4) return;
    int i = t >> 2, h = t & 3;
    const float* fp = feat + (size_t)i * (H4 * dh) + h * dh;
    const float* ap = aw + h * dh;
    float s = 0.f;
    for (int d = 0; d < dh; ++d) s += fp[d] * ap[d];
    out[t] = s;
}

__global__ void edge_max_kernel(const int* __restrict__ src, const int* __restrict__ dst,
                                const float* __restrict__ el, const float* __restrict__ er,
                                float* __restrict__ eedge, float* __restrict__ emax, int E)
{
    int t = blockIdx.x * blockDim.x + threadIdx.x;
    if (t >= E * H4) return;
    int e = t >> 2, h = t & 3;
    float v = el[src[e] * H4 + h] + er[dst[e] * H4 + h];
    v = v >= 0.f ? v : 0.2f * v;            // leaky relu, slope 0.2
    eedge[t] = v;
    atomicMaxF32(&emax[dst[e] * H4 + h], v);
}

__global__ void edge_exp_kernel(const int* __restrict__ dst,
                                float* __restrict__ eedge,
                                const float* __restrict__ emax,
                                float* __restrict__ den, int E)
{
    int t = blockIdx.x * blockDim.x + threadIdx.x;
    if (t >= E * H4) return;
    int e = t >> 2, h = t & 3;
    float ex = __expf(eedge[t] - emax[dst[e] * H4 + h]);
    eedge[t] = ex;
    atomicAdd(&den[dst[e] * H4 + h], ex);
}

__global__ void edge_agg_kernel(const int* __restrict__ src, const int* __restrict__ dst,
                                const float* __restrict__ eedge, const float* __restrict__ den,
                                const float* __restrict__ fs, float* __restrict__ rst,
                                int E, int dh)
{
    int t = blockIdx.x * blockDim.x + threadIdx.x;
    if (t >= E * H4) return;
    int e = t >> 2, h = t & 3;
    int d_ = dst[e], s_ = src[e];
    float w = eedge[t] / den[d_ * H4 + h];
    const float* fsp = fs + (size_t)s_ * (H4 * dh) + h * dh;
    float* rp = rst + (size_t)d_ * (H4 * dh) + h * dh;
    for (int d = 0; d < dh; ++d) atomicAdd(&rp[d], fsp[d] * w);
}

// rst = relu(rst + resid + bias)
__global__ void finalize_kernel(float* __restrict__ rst, const float* __restrict__ resid,
                                const float* __restrict__ bias, int n, int F)
{
    int t = blockIdx.x * blockDim.x + threadIdx.x;
    if (t >= n * F) return;
    int j = t % F;
    float v = rst[t] + resid[t] + bias[j];
    rst[t] = v > 0.f ? v : 0.f;
}

__global__ void concat2_kernel(const float* __restrict__ a, int Fa,
                               const float* __restrict__ b, int Fb,
                               float* __restrict__ out, int n)
{
    int F = Fa + Fb;
    int t = blockIdx.x * blockDim.x + threadIdx.x;
    if (t >= n * F) return;
    int i = t / F, j = t % F;
    out[t] = (j < Fa) ? a[(size_t)i * Fa + j] : b[(size_t)i * Fb + (j - Fa)];
}

__global__ void degree_kernel(const int* __restrict__ cdst, float* __restrict__ deg, int E)
{
    int t = blockIdx.x * blockDim.x + threadIdx.x;
    if (t < E) atomicAdd(&deg[cdst[t]], 1.0f);
}

// y[dst] += m[src], 8 floats per thread
__global__ void scatter_msg_kernel(const int* __restrict__ csrc, const int* __restrict__ cdst,
                                   const float* __restrict__ m, float* __restrict__ y, int E)
{
    int t = blockIdx.x * blockDim.x + threadIdx.x;
    if (t >= E * 8) return;
    int e = t >> 3, c = (t & 7) * 8;
    const float* mp = m + (size_t)csrc[e] * MSG + c;
    float* yp = y + (size_t)cdst[e] * MSG + c;
    for (int k = 0; k < 8; ++k) atomicAdd(&yp[k], mp[k]);
}

__global__ void scale_msg_kernel(float* __restrict__ y, const float* __restrict__ deg, int n)
{
    int t = blockIdx.x * blockDim.x + threadIdx.x;
    if (t >= n * MSG) return;
    int i = t / MSG;
    float d = deg[i];
    y[t] /= (d > 1.0f ? d : 1.0f);
}

__global__ void gru_kernel(const float* __restrict__ gi, const float* __restrict__ gh,
                           const float* __restrict__ state, float* __restrict__ out, int n)
{
    int t = blockIdx.x * blockDim.x + threadIdx.x;
    if (t >= n * COMM) return;
    int i = t >> 7, j = t & 127;
    const float* gip = gi + (size_t)i * (3 * COMM);
    const float* ghp = gh + (size_t)i * (3 * COMM);
    float r  = 1.f / (1.f + __expf(-(gip[j] + ghp[j])));
    float zg = 1.f / (1.f + __expf(-(gip[COMM + j] + ghp[COMM + j])));
    float nn = tanhf(gip[2 * COMM + j] + r * ghp[2 * COMM + j]);
    out[t] = (1.f - zg) * nn + zg * state[t];
}

__global__ void out_proj_kernel(const float* __restrict__ h, const float* __restrict__ w,
                                const float* __restrict__ b, float* __restrict__ out, int n)
{
    int i = blockIdx.x * blockDim.x + threadIdx.x;
    if (i >= n) return;
    float acc[OUTD];
#pragma unroll
    for (int j = 0; j < OUTD; ++j) acc[j] = b[j];
    const float* hp = h + (size_t)i * COMM;
    for (int k = 0; k < COMM; ++k) {
        float hv = hp[k];
#pragma unroll
        for (int j = 0; j < OUTD; ++j) acc[j] += hv * w[k * OUTD + j];
    }
#pragma unroll
    for (int j = 0; j < OUTD; ++j) out[(size_t)i * OUTD + j] = acc[j];
}

__global__ void copy_kernel(const float* __restrict__ src, float* __restrict__ dst, int n)
{
    int t = blockIdx.x * blockDim.x + threadIdx.x;
    if (t < n) dst[t] = src[t];
}

// ---------------------------------------------------------------------------
// Host-side launcher
// ---------------------------------------------------------------------------
static void gemm(hipStream_t s, const float* A, int lda, const float* B, int ldb,
                 float* C, int ldc, int M, int N, int K,
                 const float* bias, int acc, int relu)
{
    dim3 g(cdiv(M, 64), N / 64);
    gemm_wmma_f32<<<g, 128, 0, s>>>(A, lda, B, ldb, C, ldc, M, N, K, bias, acc, relu);
}

static void fill(hipStream_t s, float* p, float v, int n)
{
    fill_kernel<<<cdiv(n, 256), 256, 0, s>>>(p, v, n);
}

extern "C" void kernel_launch(void* const* d_in, const int* in_sizes, int n_in,
                              void* d_out, int out_size, void* d_ws, size_t ws_size,
                              hipStream_t stream)
{
    const float* feat_gt = (const float*)d_in[0];
    const float* feat_ag = (const float*)d_in[1];
    const float* z_in    = (const float*)d_in[2];
    const int*   gt_src  = (const int*)d_in[3];
    const int*   gt_dst  = (const int*)d_in[4];
    const int*   c_src   = (const int*)d_in[5];
    const int*   c_dst   = (const int*)d_in[6];
    const float* w1_src = (const float*)d_in[7];
    const float* w1_dst = (const float*)d_in[8];
    const float* a1_l   = (const float*)d_in[9];
    const float* a1_r   = (const float*)d_in[10];
    const float* b1     = (const float*)d_in[11];
    const float* w2_src = (const float*)d_in[12];
    const float* w2_dst = (const float*)d_in[13];
    const float* a2_l   = (const float*)d_in[14];
    const float* a2_r   = (const float*)d_in[15];
    const float* b2     = (const float*)d_in[16];
    const float* enc_w1 = (const float*)d_in[17];
    const float* enc_b1 = (const float*)d_in[18];
    const float* enc_w2 = (const float*)d_in[19];
    const float* enc_b2 = (const float*)d_in[20];
    const float* gru_w_ih = (const float*)d_in[21];
    const float* gru_w_hh = (const float*)d_in[22];
    const float* gru_b_ih = (const float*)d_in[23];
    const float* gru_b_hh = (const float*)d_in[24];
    const float* out_w  = (const float*)d_in[25];
    const float* out_b  = (const float*)d_in[26];

    float* out = (float*)d_out;
    float* ws  = (float*)d_ws;

    // Workspace layout (floats)
    size_t o = 0;
    float* FS   = ws + o; o += (size_t)N_GT * COMM;   // fs1 (ld VIS) / fs2 (ld COMM) reuse
    float* FD   = ws + o; o += (size_t)N_AG * COMM;   // fd1 / fd2 reuse
    float* DST2 = ws + o; o += (size_t)N_AG * COMM;
    float* EED  = ws + o; o += (size_t)E_GT * H4;     // per-edge e / exp(e), reused per layer
    float* EL   = ws + o; o += (size_t)N_GT * H4;
    float* ER   = ws + o; o += (size_t)N_AG * H4;
    float* EMAX = ws + o; o += (size_t)N_AG * H4;
    float* DEN  = ws + o; o += (size_t)N_AG * H4;
    float* HV1  = ws + o; o += (size_t)N_AG * VIS;    // layer-1 output (also its rst accum)
    float* HA   = ws + o; o += (size_t)N_AG * COMM;   // layer-2 output / comm h
    float* HB   = ws + o; o += (size_t)N_AG * COMM;   // comm h ping-pong
    float* T1   = ws + o; o += (size_t)N_AG * COMM;   // encoder hidden
    float* MM   = ws + o; o += (size_t)N_AG * MSG;    // messages
    float* YB   = ws + o; o += (size_t)N_AG * MSG;    // mailbox mean
    float* DEG  = ws + o; o += (size_t)N_AG;
    float* GI   = ws + o; o += (size_t)N_AG * 3 * COMM;
    float* GH   = ws + o; o += (size_t)N_AG * 3 * COMM;
    (void)o; (void)ws_size; (void)in_sizes; (void)n_in; (void)out_size;

    const int EB = 256;
    const int dh1 = VIS / H4;    // 16
    const int dh2 = COMM / H4;   // 32

    // ================= GAT layer 1 (gt -> agent, dh=16) =================
    gemm(stream, feat_gt, D_GT, w1_src, VIS, FS, VIS, N_GT, VIS, D_GT, nullptr, 0, 0); // fs1
    gemm(stream, feat_ag, D_AG, w1_dst, VIS, FD, VIS, N_AG, VIS, D_AG, nullptr, 0, 0); // fd1
    rowdot_kernel<<<cdiv(N_GT * H4, EB), EB, 0, stream>>>(FS, a1_l, EL, N_GT, dh1);
    rowdot_kernel<<<cdiv(N_AG * H4, EB), EB, 0, stream>>>(FD, a1_r, ER, N_AG, dh1);
    fill(stream, EMAX, -3.0e38f, N_AG * H4);
    fill(stream, DEN, 0.f, N_AG * H4);
    fill(stream, HV1, 0.f, N_AG * VIS);
    edge_max_kernel<<<cdiv(E_GT * H4, EB), EB, 0, stream>>>(gt_src, gt_dst, EL, ER, EED, EMAX, E_GT);
    edge_exp_kernel<<<cdiv(E_GT * H4, EB), EB, 0, stream>>>(gt_dst, EED, EMAX, DEN, E_GT);
    edge_agg_kernel<<<cdiv(E_GT * H4, EB), EB, 0, stream>>>(gt_src, gt_dst, EED, DEN, FS, HV1, E_GT, dh1);
    finalize_kernel<<<cdiv(N_AG * VIS, EB), EB, 0, stream>>>(HV1, feat_ag, b1, N_AG, VIS);

    // ================= GAT layer 2 (gt -> agent, dh=32) =================
    concat2_kernel<<<cdiv(N_AG * COMM, EB), EB, 0, stream>>>(HV1, VIS, feat_ag, D_AG, DST2, N_AG);
    gemm(stream, feat_gt, D_GT, w2_src, COMM, FS, COMM, N_GT, COMM, D_GT, nullptr, 0, 0); // fs2
    gemm(stream, DST2, COMM, w2_dst, COMM, FD, COMM, N_AG, COMM, COMM, nullptr, 0, 0);    // fd2
    rowdot_kernel<<<cdiv(N_GT * H4, EB), EB, 0, stream>>>(FS, a2_l, EL, N_GT, dh2);
    rowdot_kernel<<<cdiv(N_AG * H4, EB), EB, 0, stream>>>(FD, a2_r, ER, N_AG, dh2);
    fill(stream, EMAX, -3.0e38f, N_AG * H4);
    fill(stream, DEN, 0.f, N_AG * H4);
    fill(stream, HA, 0.f, N_AG * COMM);
    edge_max_kernel<<<cdiv(E_GT * H4, EB), EB, 0, stream>>>(gt_src, gt_dst, EL, ER, EED, EMAX, E_GT);
    edge_exp_kernel<<<cdiv(E_GT * H4, EB), EB, 0, stream>>>(gt_dst, EED, EMAX, DEN, E_GT);
    edge_agg_kernel<<<cdiv(E_GT * H4, EB), EB, 0, stream>>>(gt_src, gt_dst, EED, DEN, FS, HA, E_GT, dh2);
    finalize_kernel<<<cdiv(N_AG * COMM, EB), EB, 0, stream>>>(HA, DST2, b2, N_AG, COMM);

    // ================= Communication loop (2 steps) =================
    fill(stream, DEG, 0.f, N_AG);
    degree_kernel<<<cdiv(E_COMM, EB), EB, 0, stream>>>(c_dst, DEG, E_COMM);

    for (int step = 0; step < 2; ++step) {
        const float* hin   = (step == 0) ? HA : HB;
        const float* state = (step == 0) ? z_in : HB;

        // f_enc: m = relu(relu(h @ enc_w1 + b) @ enc_w2 + b)
        gemm(stream, hin, COMM, enc_w1, COMM, T1, COMM, N_AG, COMM, COMM, enc_b1, 0, 1);
        gemm(stream, T1, COMM, enc_w2, MSG, MM, MSG, N_AG, MSG, COMM, enc_b2, 0, 1);

        // mailbox mean
        fill(stream, YB, 0.f, N_AG * MSG);
        scatter_msg_kernel<<<cdiv(E_COMM * 8, EB), EB, 0, stream>>>(c_src, c_dst, MM, YB, E_COMM);
        scale_msg_kernel<<<cdiv(N_AG * MSG, EB), EB, 0, stream>>>(YB, DEG, N_AG);

        // gi = [h, y] @ w_ih + b_ih  (split GEMM, accumulate);  gh = state @ w_hh + b_hh
        gemm(stream, hin, COMM, gru_w_ih, 3 * COMM, GI, 3 * COMM, N_AG, 3 * COMM, COMM, gru_b_ih, 0, 0);
        gemm(stream, YB, MSG, gru_w_ih + (size_t)COMM * 3 * COMM, 3 * COMM, GI, 3 * COMM,
             N_AG, 3 * COMM, MSG, nullptr, 1, 0);
        gemm(stream, state, COMM, gru_w_hh, 3 * COMM, GH, 3 * COMM, N_AG, 3 * COMM, COMM, gru_b_hh, 0, 0);

        gru_kernel<<<cdiv(N_AG * COMM, EB), EB, 0, stream>>>(GI, GH, state, HB, N_AG);
    }

    // ================= Outputs: (h_out [N_AG,5], zz [N_AG,128]) flat =================
    out_proj_kernel<<<cdiv(N_AG, EB), EB, 0, stream>>>(HB, out_w, out_b, out, N_AG);
    copy_kernel<<<cdiv(N_AG * COMM, EB), EB, 0, stream>>>(HB, out + (size_t)N_AG * OUTD, N_AG * COMM);
}